// VAE_crowds_86895778333433
// MI455X (gfx1250) — compile-verified
//
#include <hip/hip_runtime.h>

typedef __attribute__((ext_vector_type(2))) float v2f;
typedef __attribute__((ext_vector_type(8))) float v8f;

constexpr int TASKS   = 100000;
constexpr int WORKERS = 500;
constexpr int FEAT    = 512;
constexpr int CLS     = 10;
constexpr int HP      = 12;                // padded row stride for h arrays (48B, 16-aligned)
constexpr int ANS     = 1000000;
constexpr int HID     = 128;
constexpr int NODES   = TASKS + WORKERS;   // 100500
constexpr int CHUNK   = 4096;              // answers per block in edge-aggregation

// ---------------- degree / normalization ----------------

__global__ void k_deg_init(int* __restrict__ deg) {
    int i = blockIdx.x * 256 + threadIdx.x;
    if (i < NODES) deg[i] = 1;   // self-loop
}

__global__ void k_deg_count(const int* __restrict__ answers, int* __restrict__ deg) {
    int a = blockIdx.x * 256 + threadIdx.x;
    if (a < ANS) {
        atomicAdd(&deg[answers[3 * a]], 1);
        atomicAdd(&deg[TASKS + answers[3 * a + 1]], 1);
    }
}

__global__ void k_dinv(const int* __restrict__ deg, float* __restrict__ dinv) {
    int i = blockIdx.x * 256 + threadIdx.x;
    if (i < NODES) dinv[i] = rsqrtf((float)deg[i]);
}

// ---------------- W_efc repack: Bp[k/2][n] = {B[k][n], B[k+1][n]} ----------------
// Makes every B fragment in the WMMA loop a single aligned 8-byte load.

__global__ void k_packB(const float* __restrict__ B, float2* __restrict__ Bp) {
    int idx = blockIdx.x * 256 + threadIdx.x;
    if (idx < (FEAT / 2) * HID) {
        int kp = idx / HID;
        int n  = idx - kp * HID;
        Bp[idx] = make_float2(B[(2 * kp) * HID + n], B[(2 * kp + 1) * HID + n]);
    }
}

// ---------------- tf_h = relu(task_feature @ W_efc + b_efc)  (f32 WMMA) ----------------
// Wave computes a 32-row x 128-col tile: 2 row-slabs x 8 col-tiles, so each B
// fragment feeds two WMMAs (register blocking halves B-load issue per matrix op).
// A 16x4 f32 frag: lanes 0-15 -> M=lane, K={k,k+1}; lanes 16-31 -> M=lane-16, K={k+2,k+3}.
// B 4x16 f32 frag: lanes 0-15 -> rows K=k,k+1; lanes 16-31 -> rows K=k+2,k+3.
// C/D 16x16 f32: vgpr j, lanes 0-15 -> (M=j, N=lane); lanes 16-31 -> (M=j+8, N=lane-16).

__global__ __launch_bounds__(256) void k_gemm_efc(const float* __restrict__ A,
                                                  const float2* __restrict__ Bp,
                                                  const float* __restrict__ bias,
                                                  float* __restrict__ out) {
    const int wave   = threadIdx.x >> 5;
    const int lane   = threadIdx.x & 31;
    const int lane16 = lane & 15;
    const int hi     = lane >> 4;

    const int slab     = blockIdx.x * 8 + wave;          // 32-row super-slab index
    const bool valid   = (slab * 32) < TASKS;
    const int row_base = valid ? slab * 32 : 0;

    const float*  arow0 = A + (size_t)(row_base + lane16) * FEAT + 2 * hi;
    const float*  arow1 = arow0 + (size_t)16 * FEAT;
    const float2* bbase = Bp + lane16;

    v8f acc[2][8];
#pragma unroll
    for (int t = 0; t < 8; ++t) {
        float bv = bias[t * 16 + lane16];          // column bias, same for all rows
        acc[0][t] = (v8f){bv, bv, bv, bv, bv, bv, bv, bv};
        acc[1][t] = acc[0][t];
    }

    for (int k = 0; k < FEAT; k += 4) {
        const float2 av0 = *(const float2*)(arow0 + k);
        const float2 av1 = *(const float2*)(arow1 + k);
        v2f af0 = {av0.x, av0.y};
        v2f af1 = {av1.x, av1.y};
        const float2* brow = bbase + (size_t)((k >> 1) + hi) * HID;
#pragma unroll
        for (int t = 0; t < 8; ++t) {
            const float2 bv = brow[t * 16];
            v2f bf = {bv.x, bv.y};
            acc[0][t] = __builtin_amdgcn_wmma_f32_16x16x4_f32(
                false, af0, false, bf, (short)0, acc[0][t], false, false);
            acc[1][t] = __builtin_amdgcn_wmma_f32_16x16x4_f32(
                false, af1, false, bf, (short)0, acc[1][t], false, false);
        }
    }

    if (valid) {
#pragma unroll
        for (int r = 0; r < 2; ++r) {
#pragma unroll
            for (int t = 0; t < 8; ++t) {
#pragma unroll
                for (int j = 0; j < 8; ++j) {
                    int m   = row_base + r * 16 + j + 8 * hi;
                    float v = acc[r][t][j];
                    out[(size_t)m * HID + t * 16 + lane16] = v > 0.f ? v : 0.f;
                }
            }
        }
    }
}

// ---------------- h_mean / h_log_std = x @ W  (f32 WMMA, N padded 10->16) ----------------

__global__ __launch_bounds__(256) void k_h(const float* __restrict__ tf_h,
                                           const float* __restrict__ wf,
                                           const float* __restrict__ Wm,
                                           const float* __restrict__ Wl,
                                           float* __restrict__ hm,
                                           float* __restrict__ hl) {
    const int wave   = threadIdx.x >> 5;
    const int lane   = threadIdx.x & 31;
    const int lane16 = lane & 15;
    const int hi     = lane >> 4;

    const int node_base = (blockIdx.x * 8 + wave) * 16;
    int nd = node_base + lane16;
    if (nd > NODES - 1) nd = NODES - 1;
    const float* xrow = (nd < TASKS) ? (tf_h + (size_t)nd * HID)
                                     : (wf + (size_t)(nd - TASKS) * HID);
    const int n = lane16;
    const bool ncol = n < CLS;

    v8f accM = {0.f, 0.f, 0.f, 0.f, 0.f, 0.f, 0.f, 0.f};
    v8f accL = {0.f, 0.f, 0.f, 0.f, 0.f, 0.f, 0.f, 0.f};

    for (int k = 0; k < HID; k += 4) {
        const int k0 = k + 2 * hi;
        const float2 av = *(const float2*)(xrow + k0);
        v2f af = {av.x, av.y};
        v2f bm, bl;
        bm.x = ncol ? Wm[k0 * CLS + n] : 0.f;
        bm.y = ncol ? Wm[(k0 + 1) * CLS + n] : 0.f;
        bl.x = ncol ? Wl[k0 * CLS + n] : 0.f;
        bl.y = ncol ? Wl[(k0 + 1) * CLS + n] : 0.f;
        accM = __builtin_amdgcn_wmma_f32_16x16x4_f32(false, af, false, bm, (short)0, accM, false, false);
        accL = __builtin_amdgcn_wmma_f32_16x16x4_f32(false, af, false, bl, (short)0, accL, false, false);
    }

    if (ncol) {
#pragma unroll
        for (int j = 0; j < 8; ++j) {
            int m = node_base + j + 8 * hi;
            if (m < NODES) {
                hm[(size_t)m * HP + n] = accM[j];
                hl[(size_t)m * HP + n] = accL[j];
            }
        }
    }
}

// ---------------- aggregation: init = bias + self-loop ----------------

__global__ void k_agg_init(const float* __restrict__ hm, const float* __restrict__ hl,
                           const float* __restrict__ dinv,
                           const float* __restrict__ bm, const float* __restrict__ bl,
                           float* __restrict__ mean, float* __restrict__ lsd) {
    int idx = blockIdx.x * 256 + threadIdx.x;
    if (idx < NODES * CLS) {
        int i = idx / CLS;
        int c = idx - i * CLS;
        float d  = dinv[i];
        float sw = d * d;                       // self-loop weight
        mean[idx] = bm[c] + hm[(size_t)i * HP + c] * sw;
        lsd[idx]  = bl[c] + hl[(size_t)i * HP + c] * sw;
    }
}

// ---------------- aggregation: bidirectional edges ----------------
// Task side: direct global f32 atomics (~10 per task row, low contention).
// Worker side: LDS accumulation per block (500x10x2 floats = 40 KB), then flush.

__device__ __forceinline__ void load_row10(const float* __restrict__ p, float* r) {
    const float4 a = *(const float4*)(p);        // rows are 48B-aligned (HP=12)
    const float4 b = *(const float4*)(p + 4);
    const float2 c = *(const float2*)(p + 8);
    r[0] = a.x; r[1] = a.y; r[2] = a.z; r[3] = a.w;
    r[4] = b.x; r[5] = b.y; r[6] = b.z; r[7] = b.w;
    r[8] = c.x; r[9] = c.y;
}

__global__ __launch_bounds__(256) void k_agg_edges(const int* __restrict__ answers,
                                                   const float* __restrict__ hm,
                                                   const float* __restrict__ hl,
                                                   const float* __restrict__ dinv,
                                                   float* __restrict__ mean,
                                                   float* __restrict__ lsd) {
    __shared__ float wacc[2 * WORKERS * CLS];
    for (int i = threadIdx.x; i < 2 * WORKERS * CLS; i += 256) wacc[i] = 0.f;
    __syncthreads();

    const int start = blockIdx.x * CHUNK;
    int end = start + CHUNK;
    if (end > ANS) end = ANS;

    for (int a = start + threadIdx.x; a < end; a += 256) {
        const int t    = answers[3 * a];
        const int w    = answers[3 * a + 1];
        const int wrow = TASKS + w;
        const float we = dinv[t] * dinv[wrow];

        float hmt[10], hmw[10], hlt[10], hlw[10];
        load_row10(hm + (size_t)t * HP,    hmt);
        load_row10(hm + (size_t)wrow * HP, hmw);
        load_row10(hl + (size_t)t * HP,    hlt);
        load_row10(hl + (size_t)wrow * HP, hlw);

        float* mt = mean + (size_t)t * CLS;
        float* lt = lsd + (size_t)t * CLS;
#pragma unroll
        for (int c = 0; c < CLS; ++c) {
            atomicAdd(&mt[c], hmw[c] * we);                              // worker -> task
            atomicAdd(&lt[c], hlw[c] * we);
            atomicAdd(&wacc[w * CLS + c], hmt[c] * we);                  // task -> worker (LDS)
            atomicAdd(&wacc[WORKERS * CLS + w * CLS + c], hlt[c] * we);
        }
    }
    __syncthreads();

    for (int i = threadIdx.x; i < WORKERS * CLS; i += 256) {
        atomicAdd(&mean[(size_t)TASKS * CLS + i], wacc[i]);
        atomicAdd(&lsd[(size_t)TASKS * CLS + i], wacc[WORKERS * CLS + i]);
    }
}

// ---------------- reparameterization + crowd output ----------------

__global__ void k_z(const float* __restrict__ mean, const float* __restrict__ lsd,
                    const float* __restrict__ eps, float* __restrict__ z) {
    int idx = blockIdx.x * 256 + threadIdx.x;
    if (idx < NODES * CLS)
        z[idx] = mean[idx] + (eps[idx] * 0.01f) * expf(lsd[idx]);
}

__global__ void k_crowd(const int* __restrict__ answers, const float* __restrict__ z,
                        float* __restrict__ crowd) {
    int idx = blockIdx.x * 256 + threadIdx.x;
    if (idx < ANS * CLS) {
        int a = idx / CLS;
        int c = idx - a * CLS;
        int t = answers[3 * a];
        int w = answers[3 * a + 1];
        crowd[idx] = z[(size_t)t * CLS + c] * z[(size_t)(TASKS + w) * CLS + c];
    }
}

// ---------------- launcher ----------------

extern "C" void kernel_launch(void* const* d_in, const int* in_sizes, int n_in,
                              void* d_out, int out_size, void* d_ws, size_t ws_size,
                              hipStream_t stream) {
    (void)in_sizes; (void)n_in; (void)out_size; (void)ws_size;

    const float* task_feature   = (const float*)d_in[0];
    const int*   answers        = (const int*)d_in[1];
    const float* worker_feature = (const float*)d_in[2];
    const float* W_efc          = (const float*)d_in[3];
    const float* b_efc          = (const float*)d_in[4];
    const float* W_mean         = (const float*)d_in[5];
    const float* b_mean         = (const float*)d_in[6];
    const float* W_ls           = (const float*)d_in[7];
    const float* b_ls           = (const float*)d_in[8];
    const float* eps            = (const float*)d_in[9];

    float* out   = (float*)d_out;
    float* crowd = out;                                   // [ANS, CLS]
    float* mean  = out + (size_t)ANS * CLS;               // [NODES, CLS]
    float* lsd   = mean + (size_t)NODES * CLS;            // [NODES, CLS]

    float*  ws   = (float*)d_ws;
    float*  tf_h = ws;                                    // [TASKS, HID]
    float2* Bp   = (float2*)(tf_h + (size_t)TASKS * HID); // [FEAT/2, HID] pairs
    float*  hm   = (float*)(Bp + (size_t)(FEAT / 2) * HID); // [NODES, HP]
    float*  hl   = hm + (size_t)NODES * HP;               // [NODES, HP]
    float*  zz   = hl + (size_t)NODES * HP;               // [NODES, CLS]
    float*  dinv = zz + (size_t)NODES * CLS;              // [NODES]
    int*    deg  = (int*)(dinv + NODES);                  // [NODES]

    auto cdiv = [](int a, int b) { return (a + b - 1) / b; };

    k_deg_init <<<cdiv(NODES, 256), 256, 0, stream>>>(deg);
    k_deg_count<<<cdiv(ANS, 256), 256, 0, stream>>>(answers, deg);
    k_dinv     <<<cdiv(NODES, 256), 256, 0, stream>>>(deg, dinv);

    k_packB    <<<cdiv((FEAT / 2) * HID, 256), 256, 0, stream>>>(W_efc, Bp);
    k_gemm_efc <<<cdiv(TASKS / 32, 8), 256, 0, stream>>>(task_feature, Bp, b_efc, tf_h);
    k_h        <<<cdiv(cdiv(NODES, 16), 8), 256, 0, stream>>>(tf_h, worker_feature,
                                                              W_mean, W_ls, hm, hl);

    k_agg_init <<<cdiv(NODES * CLS, 256), 256, 0, stream>>>(hm, hl, dinv, b_mean, b_ls, mean, lsd);
    k_agg_edges<<<cdiv(ANS, CHUNK), 256, 0, stream>>>(answers, hm, hl, dinv, mean, lsd);

    k_z        <<<cdiv(NODES * CLS, 256), 256, 0, stream>>>(mean, lsd, eps, zz);
    k_crowd    <<<cdiv(ANS * CLS, 256), 256, 0, stream>>>(answers, zz, crowd);
}